// ParsingRelationLines_69234872811937
// MI455X (gfx1250) — compile-verified
//
#include <hip/hip_runtime.h>
#include <hip/hip_bf16.h>
#include <math.h>

// Problem constants (reference: N=1024, DIM=201, R=72, C=4, G=DIM-1=200)
#define DIMC   201
#define GCH    200
#define RROWS  72
#define CLANES 4
#define POSN   (RROWS * CLANES)   // 288 positions per image

typedef float v2f __attribute__((ext_vector_type(2)));
typedef float v8f __attribute__((ext_vector_type(8)));

__device__ __forceinline__ float seg_mask(int r, int t, int d) {
    return (r >= t && r <= d) ? 1.0f : 0.0f;
}

__global__ __launch_bounds__(POSN)
void pr_lane_loss_kernel(const float* __restrict__ x,
                         const int*   __restrict__ labels,
                         float* __restrict__ lossOut) {
    const int n   = blockIdx.x;
    const int tid = threadIdx.x;            // 0..287 ; r = tid/4, c = tid%4

    __shared__ float         posS[POSN];
    __shared__ unsigned char validS[POSN];
    __shared__ float         ddS[RROWS * 2];
    __shared__ int           topS[CLANES], downS[CLANES];
    __shared__ int           segS[4];       // tL, dL, tR, dR
    __shared__ float         resS[2];       // sL, sR

    // ---------------- Phase 1: streaming online softmax expectation ----------
    // x[n, g, r, c]: for fixed (n,g) the 288 positions are contiguous ->
    // each wave load is one coalesced 128B line; x is read exactly once.
    // Branchless online rescale (2 exps/elem): under wave32 the "max updated"
    // branch would be taken wave-wide ~32/g of the time anyway, so exec-mask
    // churn costs more than the second v_exp_f32. NT hint: 237MB streamed once
    // (> 192MB L2), so don't rinse L2 with it.
    const float* xp = x + (size_t)n * DIMC * POSN + tid;
    float m = -INFINITY, s = 0.0f, t = 0.0f;
#pragma unroll 8
    for (int g = 0; g < GCH; ++g) {
        float v  = __builtin_nontemporal_load(xp + (size_t)g * POSN);
        float nm = fmaxf(m, v);
        float sc = __expf(m - nm);          // == 1 when max unchanged
        float e  = __expf(v - nm);
        s = s * sc + e;
        t = t * sc + e * (float)g;
        m = nm;
    }
    posS[tid] = t / s;                      // soft-argmax position

    int lab = labels[(size_t)n * POSN + tid];
    validS[tid] = (lab < GCH) ? 1 : 0;
    __syncthreads();

    // ---------------- Phase 2a: per-column top/down parse (4 threads) --------
    if (tid < CLANES) {
        const int c = tid;
        int first_valid = -1, first_trans = -1;
        bool vprev = false;
        for (int r = 0; r < RROWS; ++r) {
            bool v = validS[r * CLANES + c] != 0;
            if (v && first_valid < 0) first_valid = r;
            if (r > 0 && vprev && !v && first_trans < 0) first_trans = r - 1;
            vprev = v;
        }
        const bool any_valid = (first_valid >= 0);
        const bool trans0    = (first_trans == 0);
        const bool lastv     = validS[(RROWS - 1) * CLANES + c] != 0;
        int down = trans0 ? 0
                 : (lastv ? (RROWS - 1)
                 : (first_trans >= 0 ? first_trans : 0));
        int top  = any_valid ? first_valid : 0;
        down     = any_valid ? down : 0;
        topS[c]  = top;
        downS[c] = down;
    }

    // ---------------- Phase 2b: lane second differences (72 threads) ---------
    if (tid < RROWS) {
        const int r = tid;
        float p0 = posS[r * CLANES + 0];
        float p1 = posS[r * CLANES + 1];
        float p2 = posS[r * CLANES + 2];
        float p3 = posS[r * CLANES + 3];
        ddS[r * 2 + 0] = fabsf(p0 - 2.0f * p1 + p2);
        ddS[r * 2 + 1] = fabsf(p1 - 2.0f * p2 + p3);
    }
    __syncthreads();

    if (tid == 0) {
        int tL = max(topS[0], max(topS[1], topS[2]));
        int dL = min(downS[0], min(downS[1], downS[2]));
        int tR = max(topS[1], max(topS[2], topS[3]));
        int dR = min(downS[1], min(downS[2], downS[3]));
        segS[0] = tL; segS[1] = dL; segS[2] = tR; segS[3] = dR;
    }
    __syncthreads();

    // ---------------- Phase 3: masked segment sums via WMMA (wave 0) ---------
    // sL = sum_r maskL(r)*dd[r][0], sR = sum_r maskR(r)*dd[r][1]
    // A(16x4 f32): row0 = left mask chunk, row1 = right mask chunk, rest 0.
    // B(4x16 f32): col0 = dd[.,0] chunk, col1 = dd[.,1] chunk, rest 0.
    // Accumulate D over 18 K-chunks of 4 rows -> sL = D[0][0], sR = D[1][1].
    // All operand staging is branchless: unconditional ds_load (clamped index)
    // + v_cndmask, so the wmma chain has no exec-mask save/restore around it.
    if (tid < 32) {
        const int tL = segS[0], dL = segS[1], tR = segS[2], dR = segS[3];
#if defined(__gfx1250__)
        const int lane = tid;
        const int row  = lane & 15;              // A matrix row (M)
        const int kb   = (lane >> 4) << 1;       // A: lanes 0-15 -> K0,K1 ; 16-31 -> K2,K3
        const int bK   = (lane < 16) ? 0 : 1;    // B: VGPR0 holds rows K0/K1, VGPR1 K2/K3
        const int bN   = lane & 15;              // B matrix column (N)
        const int bNc  = (bN < 2) ? bN : 0;      // clamped column for safe LDS read
        v8f acc = {};
#pragma unroll
        for (int kc = 0; kc < RROWS / 4; ++kc) {
            const int r0  = kc * 4;
            const int rA0 = r0 + kb, rA1 = rA0 + 1;
            // A operand: compute both side masks unconditionally, then select.
            float mL0 = seg_mask(rA0, tL, dL), mL1 = seg_mask(rA1, tL, dL);
            float mR0 = seg_mask(rA0, tR, dR), mR1 = seg_mask(rA1, tR, dR);
            float a0 = (row == 0) ? mL0 : ((row == 1) ? mR0 : 0.0f);
            float a1 = (row == 0) ? mL1 : ((row == 1) ? mR1 : 0.0f);
            v2f a; a[0] = a0; a[1] = a1;
            // B operand: unconditional LDS loads, zero-select for columns >= 2.
            const int rB0 = r0 + bK;             // row for B VGPR0
            const int rB1 = r0 + bK + 2;         // row for B VGPR1
            float b0v = ddS[rB0 * 2 + bNc];
            float b1v = ddS[rB1 * 2 + bNc];
            v2f b; b[0] = (bN < 2) ? b0v : 0.0f;
            b[1] = (bN < 2) ? b1v : 0.0f;
            acc = __builtin_amdgcn_wmma_f32_16x16x4_f32(
                /*neg_a=*/false, a, /*neg_b=*/false, b,
                /*c_mod=*/(short)0, acc, /*reuse_a=*/false, /*reuse_b=*/false);
        }
        if (tid == 0) resS[0] = acc[0];          // D[0][0] -> VGPR0, lane 0
        if (tid == 1) resS[1] = acc[1];          // D[1][1] -> VGPR1, lane 1
#else
        if (tid == 0) {
            float sL = 0.0f, sR = 0.0f;
            for (int r = 0; r < RROWS; ++r) {
                sL += seg_mask(r, tL, dL) * ddS[r * 2 + 0];
                sR += seg_mask(r, tR, dR) * ddS[r * 2 + 1];
            }
            resS[0] = sL; resS[1] = sR;
        }
#endif
    }
    __syncthreads();

    // ---------------- Phase 4: per-image loss (deterministic slot write) -----
    if (tid == 0) {
        const int tL = segS[0], dL = segS[1], tR = segS[2], dR = segS[3];
        const bool aL = tL < dL, aR = tR < dR;
        const float WL = (float)(dL - tL + 1);
        const float WR = (float)(dR - tR + 1);
        // npairs == 1 for both sides (C=4)
        const float lossL = aL ? resS[0] / (WL * WL) : 0.0f;
        const float wL    = aL ? WL : 0.0f;
        const float lossR = aR ? resS[1] / (WR * WR) : 0.0f;
        const float wR    = aR ? WR : 0.0f;
        const float denom = (aL && aR) ? (2.0f * (float)RROWS) : (float)RROWS;
        lossOut[n] = (lossL * wL + lossR * wR) / denom;
    }
}

// Deterministic fixed-order reduction: total / count(nonzero), no atomics.
__global__ __launch_bounds__(256)
void pr_reduce(const float* __restrict__ lossK, float* __restrict__ out, int nImg) {
    __shared__ float sSum[256];
    __shared__ int   sCnt[256];
    const int tid = threadIdx.x;
    float s = 0.0f;
    int   c = 0;
    for (int i = tid; i < nImg; i += 256) {   // fixed per-thread order
        float v = lossK[i];
        s += v;
        c += (v != 0.0f) ? 1 : 0;
    }
    sSum[tid] = s; sCnt[tid] = c;
    __syncthreads();
    for (int off = 128; off > 0; off >>= 1) { // fixed tree order
        if (tid < off) {
            sSum[tid] += sSum[tid + off];
            sCnt[tid] += sCnt[tid + off];
        }
        __syncthreads();
    }
    if (tid == 0) out[0] = (sCnt[0] > 0) ? (sSum[0] / (float)sCnt[0]) : 0.0f;
}

extern "C" void kernel_launch(void* const* d_in, const int* in_sizes, int n_in,
                              void* d_out, int out_size, void* d_ws, size_t ws_size,
                              hipStream_t stream) {
    const float* x      = (const float*)d_in[0];
    const int*   labels = (const int*)d_in[1];
    float*       out    = (float*)d_out;
    float*       ws     = (float*)d_ws;       // nImg floats of per-image loss_k

    const int nImg = in_sizes[0] / (DIMC * POSN);   // 1024 for reference shapes

    pr_lane_loss_kernel<<<nImg, POSN, 0, stream>>>(x, labels, ws);
    pr_reduce<<<1, 256, 0, stream>>>(ws, out, nImg);
}